// WeightingLayer_71768903516645
// MI455X (gfx1250) — compile-verified
//
#include <hip/hip_runtime.h>
#include <hip/hip_bf16.h>

typedef __attribute__((ext_vector_type(2))) float v2f;
typedef __attribute__((ext_vector_type(8))) float v8f;

#define C_IN   32
#define C1_CH  16
#define C2_CH  8
#define NBATCH 8

// ---------------------------------------------------------------------------
// Phase 1: per-point MLP scores via V_WMMA_F32_16X16X4_F32.
// Each wave processes 16 points per tile. Weights (A-fragments) are preloaded
// once per wave. Loads use uniform SGPR channel bases + one 32-bit VGPR
// offset per tile (GVS addressing) to avoid 64-bit VALU address math.
// ---------------------------------------------------------------------------
__global__ __launch_bounds__(256) void pointnet_score_kernel(
    const float* __restrict__ X,
    const float* __restrict__ W1, const float* __restrict__ b1,
    const float* __restrict__ W2, const float* __restrict__ b2,
    const float* __restrict__ W3, const float* __restrict__ b3,
    float* __restrict__ S, int N)
{
  const int lane = threadIdx.x & 31;
  const int half = lane >> 4;   // half-wave: selects K offset in A/B fragments
  const int l16  = lane & 15;
  const int wavesPerBlock = blockDim.x >> 5;
  const int waveId   = blockIdx.x * wavesPerBlock + (threadIdx.x >> 5);
  const int numWaves = gridDim.x * wavesPerBlock;

  // ---- A-fragment preload: W1 (16x32) as 8 chunks of 16x4 ----
  // f32 16x4 A layout: lanes0-15 -> M=l16, V0=K0,V1=K1 ; lanes16-31 -> V0=K2,V1=K3
  v2f a1[8];
#pragma unroll
  for (int kc = 0; kc < 8; ++kc) {
    const int k0 = kc * 4 + half * 2;
    a1[kc][0] = W1[l16 * C_IN + k0];
    a1[kc][1] = W1[l16 * C_IN + k0 + 1];
  }
  // ---- W2 (8x16) zero-padded to 16x16, 4 chunks of 16x4 ----
  v2f a2[4];
#pragma unroll
  for (int kc = 0; kc < 4; ++kc) {
    const int k0 = kc * 4 + half * 2;
    a2[kc][0] = (l16 < C2_CH) ? W2[l16 * C1_CH + k0]     : 0.f;
    a2[kc][1] = (l16 < C2_CH) ? W2[l16 * C1_CH + k0 + 1] : 0.f;
  }
  // Per-lane bias for the C layout (VGPR r: lanes0-15 -> M=r, lanes16-31 -> M=r+8)
  float bias1[8], bias2[8];
#pragma unroll
  for (int r = 0; r < 8; ++r) {
    const int m = r + half * 8;
    bias1[r] = b1[m];
    bias2[r] = (m < C2_CH) ? b2[m] : 0.f;
  }
  float w3[8];
#pragma unroll
  for (int m = 0; m < 8; ++m) w3[m] = W3[m];
  const float b3s = b3[0];

  const int tilesPerBatch = N >> 4;
  const int halfOff = half * 2 * N;          // lane-varying, 32-bit safe
  const int tileStride = numWaves << 4;      // points advanced per iteration

  for (int b = 0; b < NBATCH; ++b) {
    const float* __restrict__ Xb = X + (size_t)b * C_IN * N;   // uniform (SGPR)
    float* __restrict__ Sb = S + (size_t)b * N;

    for (int tt = waveId; tt < tilesPerBatch; tt += numWaves) {
      const int n0  = tt << 4;
      const int off = halfOff + n0 + l16;    // 32-bit VGPR offset (GVS form)

      // ---- layer 1: 32->16, 8 chained WMMAs (K accumulate) ----
      v8f c1 = {0.f, 0.f, 0.f, 0.f, 0.f, 0.f, 0.f, 0.f};
#pragma unroll
      for (int kc = 0; kc < 8; ++kc) {
        const float* __restrict__ Xk = Xb + (size_t)(kc * 4) * N;  // uniform base
        v2f bb;                                // B frag: 4x16 of X
        bb[0] = Xk[off];
        bb[1] = Xk[off + N];
        // prefetch one tile-iteration ahead (stream front-running into L2)
        __builtin_prefetch(&Xk[off + tileStride], 0, 1);
        __builtin_prefetch(&Xk[off + N + tileStride], 0, 1);
        c1 = __builtin_amdgcn_wmma_f32_16x16x4_f32(
                false, a1[kc], false, bb, (short)0, c1, false, false);
      }
#pragma unroll
      for (int r = 0; r < 8; ++r) c1[r] = fmaxf(c1[r] + bias1[r], 0.f);

      // ---- layer 2: 16->8 (M padded to 16), 4 WMMAs; C->B relayout via shfl ----
      v8f c2 = {0.f, 0.f, 0.f, 0.f, 0.f, 0.f, 0.f, 0.f};
#pragma unroll
      for (int kc = 0; kc < 4; ++kc) {
        const int base = (kc & 1) * 4;
        v2f bb;
        if (kc < 2) {
          // needed rows (K=4kc..4kc+3) live in lanes 0-15 of c1[base..base+3]
          const float sy0 = __shfl_xor(c1[base + 2], 16, 32);
          const float sy1 = __shfl_xor(c1[base + 3], 16, 32);
          bb[0] = half ? sy0 : c1[base];
          bb[1] = half ? sy1 : c1[base + 1];
        } else {
          // needed rows live in lanes 16-31
          const float sx0 = __shfl_xor(c1[base], 16, 32);
          const float sx1 = __shfl_xor(c1[base + 1], 16, 32);
          bb[0] = half ? c1[base + 2] : sx0;
          bb[1] = half ? c1[base + 3] : sx1;
        }
        c2 = __builtin_amdgcn_wmma_f32_16x16x4_f32(
                false, a2[kc], false, bb, (short)0, c2, false, false);
      }
#pragma unroll
      for (int r = 0; r < 8; ++r) c2[r] = fmaxf(c2[r] + bias2[r], 0.f);

      // ---- layer 3: 8->1 dot + softplus; lanes 0-15 each own one point ----
      float acc = b3s;
#pragma unroll
      for (int m = 0; m < 8; ++m) acc = fmaf(w3[m], c2[m], acc);
      const float sp = (acc > 20.f) ? acc : log1pf(expf(acc));
      if (half == 0) Sb[n0 + l16] = sp;
    }
  }
}

// ---------------------------------------------------------------------------
// Phase 2: per-batch top-K (sorted descending) via bit-histogram threshold +
// candidate collection + K parallel argmax rounds. Scores are softplus > 0,
// so raw float bits order identically to the float values as unsigned.
// ---------------------------------------------------------------------------
#define TK_BINS 2048
#define TK_CAP  4096

__global__ __launch_bounds__(1024) void topk_kernel(
    const float* __restrict__ S, int* __restrict__ out, int N, int K)
{
  __shared__ unsigned hist[TK_BINS];
  __shared__ unsigned long long ckey[TK_CAP];
  __shared__ unsigned cnt;
  __shared__ unsigned thrBin;
  __shared__ unsigned long long rkey[1024];
  __shared__ int rslot[1024];

  const int tid = threadIdx.x;
  const int b   = blockIdx.x;
  const float* Sb = S + (size_t)b * N;

  for (int i = tid; i < TK_BINS; i += blockDim.x) hist[i] = 0u;
  if (tid == 0) cnt = 0u;
  __syncthreads();

  // Pass 1: histogram on top 11 bits (sign=0 for softplus outputs)
  for (int i = tid; i < N; i += blockDim.x) {
    const unsigned bits = __float_as_uint(Sb[i]);
    atomicAdd(&hist[bits >> 20], 1u);
  }
  __syncthreads();

  if (tid == 0) {
    unsigned acc = 0; int t = TK_BINS - 1;
    for (; t >= 0; --t) { acc += hist[t]; if (acc >= (unsigned)K) break; }
    thrBin = (t < 0) ? 0u : (unsigned)t;
  }
  __syncthreads();
  const unsigned T = thrBin;

  // Pass 2: collect candidates with bin >= T
  for (int i = tid; i < N; i += blockDim.x) {
    const unsigned bits = __float_as_uint(Sb[i]);
    if ((bits >> 20) >= T) {
      const unsigned p = atomicAdd(&cnt, 1u);
      if (p < TK_CAP)
        ckey[p] = ((unsigned long long)bits << 32) | (unsigned)i;
    }
  }
  __syncthreads();
  const int m = (cnt < (unsigned)TK_CAP) ? (int)cnt : TK_CAP;

  // K rounds of parallel argmax -> sorted descending indices
  for (int r = 0; r < K; ++r) {
    unsigned long long best = 0ull; int bslot = -1;
    for (int i = tid; i < m; i += blockDim.x) {
      const unsigned long long k = ckey[i];
      if (k > best) { best = k; bslot = i; }
    }
    rkey[tid] = best; rslot[tid] = bslot;
    __syncthreads();
    for (int s = blockDim.x >> 1; s > 0; s >>= 1) {
      if (tid < s && rkey[tid + s] > rkey[tid]) {
        rkey[tid] = rkey[tid + s]; rslot[tid] = rslot[tid + s];
      }
      __syncthreads();
    }
    if (tid == 0) {
      const int slot = rslot[0];
      out[b * K + r] = (slot >= 0) ? (int)(unsigned)(rkey[0] & 0xffffffffu) : 0;
      if (slot >= 0) ckey[slot] = 0ull;   // remove winner
    }
    __syncthreads();
  }
}

// ---------------------------------------------------------------------------
// Launch. Inputs: X, K, W1, b1, W2, b2, W3, b3. Output: 8*K int32 indices.
// Workspace: scores S[8][N] (12.8 MB).
// ---------------------------------------------------------------------------
extern "C" void kernel_launch(void* const* d_in, const int* in_sizes, int n_in,
                              void* d_out, int out_size, void* d_ws, size_t ws_size,
                              hipStream_t stream) {
  const float* X  = (const float*)d_in[0];
  const float* W1 = (const float*)d_in[2];
  const float* b1 = (const float*)d_in[3];
  const float* W2 = (const float*)d_in[4];
  const float* b2 = (const float*)d_in[5];
  const float* W3 = (const float*)d_in[6];
  const float* b3 = (const float*)d_in[7];

  const int N = in_sizes[0] / (NBATCH * C_IN);   // 400000
  const int K = out_size / NBATCH;               // 64

  float* S  = (float*)d_ws;
  int* out  = (int*)d_out;

  pointnet_score_kernel<<<dim3(1024), dim3(256), 0, stream>>>(
      X, W1, b1, W2, b2, W3, b3, S, N);
  topk_kernel<<<dim3(NBATCH), dim3(1024), 0, stream>>>(S, out, N, K);
}